// CreateMultiScaleWarping_20066087207636
// MI455X (gfx1250) — compile-verified
//
#include <hip/hip_runtime.h>

#if defined(__AMDGCN__) && __has_builtin(__builtin_amdgcn_global_load_async_to_lds_b32)
#define USE_ASYNC_LDS 1
#else
#define USE_ASYNC_LDS 0
#endif

// Bilinear warp matching jax reference:
//   gx = -1 + (i-1)/(S-1)*2 + flow[b,0,i,j] ;  x = (gx+1)*0.5*(S-1)   (x indexes WIDTH)
//   gy = -1 + (j-1)/(S-1)*2 + flow[b,1,i,j] ;  y = (gy+1)*0.5*(S-1)   (y indexes HEIGHT)
//   out[b,c,i,j] = sum of 4 taps img[b,c,yc,xc] * wy*wx * inbounds
template <int S>
__global__ __launch_bounds__(256) void msw_warp_kernel(
    const float* __restrict__ img, const float* __restrict__ flow,
    float* __restrict__ out) {
  constexpr int B = 16, C = 16, SS = S * S;
  const int tid = threadIdx.x;
  const int p = blockIdx.x * 256 + tid;   // flat (b,i,j); grid divides exactly
  const int j = p & (S - 1);
  const int i = (p / S) & (S - 1);
  const int b = p / SS;

  const int fx_idx = p + b * SS;          // flow[b,0,i,j]  ([B,2,S,S] layout)
  const int fy_idx = fx_idx + SS;         // flow[b,1,i,j]

  float fx, fy;
#if USE_ASYNC_LDS
  __shared__ float sflow[2 * 256];
  {
    // Builtin signature (per hipcc diagnostic) takes generic int* args;
    // codegen inserts the global/LDS addrspace casts.
    __builtin_amdgcn_global_load_async_to_lds_b32(
        (int*)(flow + fx_idx), (int*)&sflow[tid], 0, 0);
    __builtin_amdgcn_global_load_async_to_lds_b32(
        (int*)(flow + fy_idx), (int*)&sflow[256 + tid], 0, 0);
#if __has_builtin(__builtin_amdgcn_s_wait_asynccnt)
    __builtin_amdgcn_s_wait_asynccnt(0);
#else
    asm volatile("s_wait_asynccnt 0" ::: "memory");
#endif
    asm volatile("" ::: "memory");  // keep LDS reads below the wait
    fx = sflow[tid];
    fy = sflow[256 + tid];
  }
#else
  fx = flow[fx_idx];
  fy = flow[fy_idx];
#endif

  // Speculative prefetch of flow far ahead (gfx1250 global_prefetch_b8).
  if (p + 8192 < B * SS) __builtin_prefetch(flow + fx_idx + 8192, 0, 0);

  constexpr float inv = 2.0f / (float)(S - 1);
  constexpr float half = 0.5f * (float)(S - 1);
  const float gx = -1.0f + (float)(i - 1) * inv + fx;
  const float gy = -1.0f + (float)(j - 1) * inv + fy;
  const float x = (gx + 1.0f) * half;     // width coordinate
  const float y = (gy + 1.0f) * half;     // height coordinate

  const float x0f = floorf(x), y0f = floorf(y);
  const float x1f = x0f + 1.0f, y1f = y0f + 1.0f;
  const float wx1 = x - x0f, wx0 = 1.0f - wx1;
  const float wy1 = y - y0f, wy0 = 1.0f - wy1;

  constexpr float lim = (float)(S - 1);
  const float mx0 = (x0f >= 0.0f && x0f <= lim) ? 1.0f : 0.0f;
  const float mx1 = (x1f >= 0.0f && x1f <= lim) ? 1.0f : 0.0f;
  const float my0 = (y0f >= 0.0f && y0f <= lim) ? 1.0f : 0.0f;
  const float my1 = (y1f >= 0.0f && y1f <= lim) ? 1.0f : 0.0f;

  const int x0c = (int)fminf(fmaxf(x0f, 0.0f), lim);
  const int x1c = (int)fminf(fmaxf(x1f, 0.0f), lim);
  const int y0c = (int)fminf(fmaxf(y0f, 0.0f), lim);
  const int y1c = (int)fminf(fmaxf(y1f, 0.0f), lim);

  // Fold inbounds masks into the tap weights (reference multiplies value*inb).
  const float w00 = wy0 * wx0 * (my0 * mx0);
  const float w01 = wy0 * wx1 * (my0 * mx1);
  const float w10 = wy1 * wx0 * (my1 * mx0);
  const float w11 = wy1 * wx1 * (my1 * mx1);

  const int l00 = y0c * S + x0c;
  const int l01 = y0c * S + x1c;
  const int l10 = y1c * S + x0c;
  const int l11 = y1c * S + x1c;

  const float* __restrict__ imgb = img + (long)b * C * SS;
  float* __restrict__ outb = out + (long)b * C * SS + (i * S + j);

#pragma unroll
  for (int c = 0; c < C; ++c) {
    const float* __restrict__ pl = imgb + c * SS;
    const float v00 = pl[l00];
    const float v01 = pl[l01];
    const float v10 = pl[l10];
    const float v11 = pl[l11];
    const float r = fmaf(v00, w00, fmaf(v01, w01, fmaf(v10, w10, v11 * w11)));
    // Output is streamed, never re-read: non-temporal store preserves L2 for img.
    __builtin_nontemporal_store(r, outb + c * SS);
  }
}

extern "C" void kernel_launch(void* const* d_in, const int* in_sizes, int n_in,
                              void* d_out, int out_size, void* d_ws, size_t ws_size,
                              hipStream_t stream) {
  (void)in_sizes; (void)n_in; (void)d_ws; (void)ws_size; (void)out_size;
  const float* img0  = (const float*)d_in[0];
  const float* flow0 = (const float*)d_in[1];
  const float* img1  = (const float*)d_in[2];
  const float* flow1 = (const float*)d_in[3];
  const float* img2  = (const float*)d_in[4];
  const float* flow2 = (const float*)d_in[5];
  const float* img3  = (const float*)d_in[6];
  const float* flow3 = (const float*)d_in[7];
  float* out = (float*)d_out;

  constexpr long B = 16, C = 16;
  const long n0 = B * C * 512L * 512L;   // 67108864
  const long n1 = B * C * 256L * 256L;   // 16777216
  const long n2 = B * C * 128L * 128L;   //  4194304
  const long n3 = B * C *  64L *  64L;   //  1048576
  (void)n3;

  // One thread per (b,i,j): blocks = B*S*S/256 (exact).
  msw_warp_kernel<512><<<16 * 512 * 512 / 256, 256, 0, stream>>>(img0, flow0, out);
  msw_warp_kernel<256><<<16 * 256 * 256 / 256, 256, 0, stream>>>(img1, flow1, out + n0);
  msw_warp_kernel<128><<<16 * 128 * 128 / 256, 256, 0, stream>>>(img2, flow2, out + n0 + n1);
  msw_warp_kernel< 64><<<16 *  64 *  64 / 256, 256, 0, stream>>>(img3, flow3, out + n0 + n1 + n2);
}